// Net_59021440581796
// MI455X (gfx1250) — compile-verified
//
#include <hip/hip_runtime.h>
#include <math.h>

#define NN    100000
#define EE    1600000
#define FIN   100
#define CC    47
#define CPAD  48
#define MTILES (NN / 16)   // 6250, exact

typedef float v2f __attribute__((ext_vector_type(2)));
typedef float v8f __attribute__((ext_vector_type(8)));

// ---------------- degree / dinv ----------------
__global__ void k_init_deg(float* __restrict__ deg) {
  int i = blockIdx.x * blockDim.x + threadIdx.x;
  if (i < NN) deg[i] = 1.0f;                 // self-loop contributes 1
}

__global__ void k_count_deg(const int* __restrict__ dst, float* __restrict__ deg) {
  int e = blockIdx.x * blockDim.x + threadIdx.x;
  if (e < EE) atomicAdd(&deg[dst[e]], 1.0f);
}

__global__ void k_dinv(float* __restrict__ deg) {
  int i = blockIdx.x * blockDim.x + threadIdx.x;
  if (i < NN) deg[i] = rsqrtf(deg[i]);       // deg >= 1 always (self loops)
}

// ---------------- pad W ([100][47] -> [100][48]) and bias ----------------
__global__ void k_pad_wb(const float* __restrict__ W, const float* __restrict__ b,
                         float* __restrict__ Wp, float* __restrict__ bp) {
  int i = blockIdx.x * blockDim.x + threadIdx.x;
  if (i < FIN * CPAD) {
    int k = i / CPAD, c = i % CPAD;
    Wp[i] = (c < CC) ? W[k * CC + c] : 0.0f;
  }
  if (i < CPAD) bp[i] = (i < CC) ? b[i] : 0.0f;
}

// ---------------- h0 = x @ W + b via V_WMMA_F32_16X16X4_F32 ----------------
// A 16x4 f32 layout: lanes 0-15 hold row m, K = 0,1 in the 2 regs;
//                    lanes 16-31 hold row m, K = 2,3.
// B 4x16 f32 layout: lanes 0-15 hold col m, K = 0,1; lanes 16-31 K = 2,3.
// C/D 16x16 f32:     reg v, lanes 0-15 -> (row v,   col m)
//                           lanes 16-31 -> (row v+8, col m)
__global__ __launch_bounds__(256) void k_gemm_h0(
    const float* __restrict__ x, const float* __restrict__ Wp,
    const float* __restrict__ bp, float* __restrict__ h0) {
  int lane  = threadIdx.x & 31;
  int wave  = threadIdx.x >> 5;
  int mtile = blockIdx.x * 8 + wave;
  if (mtile >= MTILES) return;               // wave-uniform exit; EXEC stays all-1s
  int m    = lane & 15;
  int kh   = (lane >> 4) << 1;               // 0 or 2
  int row0 = mtile * 16;
  const float* xa = x  + (size_t)(row0 + m) * FIN + kh;
  const float* wb = Wp + (size_t)kh * CPAD + m;
  v8f acc0 = {}, acc1 = {}, acc2 = {};
#pragma unroll 5
  for (int s = 0; s < FIN / 4; ++s) {
    v2f a;  a.x  = xa[0];        a.y  = xa[1];
    v2f b0; b0.x = wb[0];        b0.y = wb[CPAD];
    v2f b1; b1.x = wb[16];       b1.y = wb[CPAD + 16];
    v2f b2; b2.x = wb[32];       b2.y = wb[CPAD + 32];
    acc0 = __builtin_amdgcn_wmma_f32_16x16x4_f32(false, a, false, b0, (short)0, acc0, false, false);
    acc1 = __builtin_amdgcn_wmma_f32_16x16x4_f32(false, a, false, b1, (short)0, acc1, false, false);
    acc2 = __builtin_amdgcn_wmma_f32_16x16x4_f32(false, a, false, b2, (short)0, acc2, false, false);
    xa += 4;
    wb += 4 * CPAD;
  }
  int   rbase = row0 + ((lane >> 4) << 3);
  float bv0 = bp[m], bv1 = bp[16 + m], bv2 = bp[32 + m];
#pragma unroll
  for (int v = 0; v < 8; ++v) {
    float* o = h0 + (size_t)(rbase + v) * CC;
    o[m]      = acc0[v] + bv0;
    o[16 + m] = acc1[v] + bv1;
    if (32 + m < CC) o[32 + m] = acc2[v] + bv2;   // col 47 is padding
  }
}

// ---------------- per-hop init: teleport + self-loop edge folded in ----------------
__global__ void k_hop_init(const float* __restrict__ h0, const float* __restrict__ hin,
                           const float* __restrict__ dinv, float* __restrict__ hout) {
  int i = blockIdx.x * blockDim.x + threadIdx.x;
  if (i < NN * CC) {
    int   n  = i / CC;
    float di = dinv[n];
    hout[i] = 0.1f * h0[i] + 0.9f * di * di * hin[i];
  }
}

// ---------------- edge scatter-add: one thread per (edge, channel) ----------------
__global__ void k_edge_prop(const int* __restrict__ src, const int* __restrict__ dst,
                            const float* __restrict__ dinv, const float* __restrict__ hin,
                            float* __restrict__ hout) {
  long long i = (long long)blockIdx.x * blockDim.x + threadIdx.x;
  if (i >= (long long)EE * CC) return;
  int e = (int)(i / CC);
  int c = (int)(i - (long long)e * CC);
  int s = src[e], d = dst[e];
  float w = 0.9f * dinv[s] * dinv[d];
  atomicAdd(&hout[(size_t)d * CC + c], w * hin[(size_t)s * CC + c]);
}

// ---------------- row-wise log_softmax ----------------
__global__ void k_logsoftmax(const float* __restrict__ h, float* __restrict__ out) {
  int n = blockIdx.x * blockDim.x + threadIdx.x;
  if (n >= NN) return;
  const float* r = h + (size_t)n * CC;
  float v[CC];
  float mx = -3.402823466e38f;
#pragma unroll
  for (int c = 0; c < CC; ++c) { v[c] = r[c]; mx = fmaxf(mx, v[c]); }
  float sum = 0.0f;
#pragma unroll
  for (int c = 0; c < CC; ++c) sum += expf(v[c] - mx);
  float lse = mx + logf(sum);
  float* o = out + (size_t)n * CC;
#pragma unroll
  for (int c = 0; c < CC; ++c) o[c] = v[c] - lse;
}

// ---------------- launch ----------------
extern "C" void kernel_launch(void* const* d_in, const int* in_sizes, int n_in,
                              void* d_out, int out_size, void* d_ws, size_t ws_size,
                              hipStream_t stream) {
  (void)in_sizes; (void)n_in; (void)out_size; (void)ws_size;
  const float* x    = (const float*)d_in[0];           // [N, F_IN]
  const int*   ei   = (const int*)d_in[1];             // [2, E]
  const float* W    = (const float*)d_in[2];           // [F_IN, C]
  const float* bias = (const float*)d_in[3];           // [C]
  const int* src = ei;
  const int* dst = ei + EE;

  // workspace carve-out (256B aligned regions)
  size_t off = 0;
  char*  base = (char*)d_ws;
  auto carve = [&](size_t bytes) -> void* {
    void* p = base + off;
    off = (off + bytes + 255) & ~(size_t)255;
    return p;
  };
  float* dinv = (float*)carve((size_t)NN * 4);
  float* Wp   = (float*)carve((size_t)FIN * CPAD * 4);
  float* bp   = (float*)carve((size_t)CPAD * 4);
  float* h0   = (float*)carve((size_t)NN * CC * 4);
  float* hA   = (float*)carve((size_t)NN * CC * 4);
  float* hB   = (float*)carve((size_t)NN * CC * 4);

  const int T = 256;
  k_init_deg <<<(NN + T - 1) / T, T, 0, stream>>>(dinv);
  k_count_deg<<<(EE + T - 1) / T, T, 0, stream>>>(dst, dinv);
  k_dinv     <<<(NN + T - 1) / T, T, 0, stream>>>(dinv);
  k_pad_wb   <<<(FIN * CPAD + T - 1) / T, T, 0, stream>>>(W, bias, Wp, bp);
  k_gemm_h0  <<<(MTILES + 7) / 8, T, 0, stream>>>(x, Wp, bp, h0);

  const long long ec = (long long)EE * CC;
  const int nc = NN * CC;
  const float* hin = h0;
  float* hout = hA;
  float* spare = hB;
  for (int hop = 0; hop < 3; ++hop) {
    k_hop_init <<<(nc + T - 1) / T, T, 0, stream>>>(h0, hin, dinv, hout);
    k_edge_prop<<<(unsigned)((ec + T - 1) / T), T, 0, stream>>>(src, dst, dinv, hin, hout);
    const float* new_in = hout;
    float* new_out = (hop == 0) ? spare : (float*)hin;  // ping-pong hA/hB (h0 preserved)
    hin = new_in;
    hout = new_out;
  }
  // after 3 hops result is in `hin`
  k_logsoftmax<<<(NN + T - 1) / T, T, 0, stream>>>(hin, (float*)d_out);
}